// GNNA1D_L_86878598464037
// MI455X (gfx1250) — compile-verified
//
#include <hip/hip_runtime.h>
#include <math.h>

// ---------------------------------------------------------------------------
// Problem constants (from reference setup_inputs): B=16, nR=4, L=100, K=8, U=2
// N_EFF = B*nR = 64 effective batch; channels padded to 64, L padded to 112.
// ---------------------------------------------------------------------------
#define N_EFF   64
#define LVALID  100
#define LP      112
#define CH      64
#define PLANE   (CH * LP)        // 7168 floats per batch element
#define BN_EPS  1e-5f

typedef __attribute__((ext_vector_type(2))) float v2f;
typedef __attribute__((ext_vector_type(8))) float v8f;
typedef __attribute__((ext_vector_type(4))) unsigned int u32x4;
typedef __attribute__((ext_vector_type(4))) int i32x4;
typedef __attribute__((ext_vector_type(8))) int i32x8;

// Tensor Data Mover path (gfx1250): guarded so the file compiles on host pass
// and on toolchains lacking the builtin.
#if defined(__gfx1250__) && defined(__AMDGCN__)
#if __has_builtin(__builtin_amdgcn_tensor_load_to_lds) && \
    __has_builtin(__builtin_amdgcn_s_wait_tensorcnt)
#define USE_TDM 1
#endif
#endif

__device__ __forceinline__ v8f wmma_f32(v2f a, v2f b, v8f c) {
    // V_WMMA_F32_16X16X4_F32 : D = A(16x4) * B(4x16) + C(16x16), all fp32
    return __builtin_amdgcn_wmma_f32_16x16x4_f32(
        /*neg_a=*/false, a, /*neg_b=*/false, b,
        /*c_mod=*/(short)0, c, /*reuse_a=*/false, /*reuse_b=*/false);
}

// ---------------------------------------------------------------------------
// Pack Hhat_{real,imag} [N,L,K,U] -> padded plane [N, 64ch, 112L].
// channel c = k*U+u (real, c<16) / 16 + k*U+u (imag). Padding = 0.
// ---------------------------------------------------------------------------
__global__ void pack_kernel(const float* __restrict__ re,
                            const float* __restrict__ im,
                            float* __restrict__ bufA) {
    int idx = blockIdx.x * 256 + threadIdx.x;
    if (idx >= N_EFF * PLANE) return;
    int n = idx / PLANE;
    int c = (idx / LP) & (CH - 1);
    int l = idx % LP;
    float v = 0.f;
    if (l < LVALID && c < 32) {
        int ku = c & 15;
        const float* src = (c < 16) ? re : im;
        v = src[(n * LVALID + l) * 16 + ku];
    }
    bufA[idx] = v;
}

__global__ void zero_stats_kernel(float* __restrict__ stats) {
    if (threadIdx.x < 2 * CH) stats[threadIdx.x] = 0.f;
}

// ---------------------------------------------------------------------------
// Fused attention layer: one workgroup (8 wave32) per batch element.
//   S = P1*A + P2*A                        (WMMA f32 16x16x4)
//   Kf = k*A ; Y[o] = sum_m |Kf|*V / ||Kf||   (rank-1 attention collapsed)
//   out = relu(S + 0.1*Y)  (+ per-channel BN sum/sumsq via global atomics)
// q-matmul is provably dead (Alpha_norm independent of Q) and skipped.
// Activation plane staged in LDS via the Tensor Data Mover when available.
// ---------------------------------------------------------------------------
__global__ __launch_bounds__(256) void layer_kernel(
    const float* __restrict__ Ain, float* __restrict__ Aout,
    const float* __restrict__ W1, const float* __restrict__ W2,
    const float* __restrict__ Wk, int din, int dout,
    float* __restrict__ gstats, int relu_bn)
{
    __shared__ float Alds[PLANE];   // input activations  [64][112]
    __shared__ float Slds[PLANE];   // A1+V               [64][112]
    __shared__ float red1[CH];      // sum |Kf|*V  -> later Y
    __shared__ float red2[CH];      // sum Kf^2

    const int tid  = threadIdx.x;
    const int n    = blockIdx.x;
    const int wave = tid >> 5;
    const int lane = tid & 31;
    const int lrow = lane & 15;
    const int g    = lane >> 4;     // half-wave group (K/M offset selector)

    // Warm L2/L0 for the (L2-resident) weight matrices: global_prefetch_b8
    if (tid * 16 < dout * din) {
        __builtin_prefetch(W1 + tid * 16, 0, 1);
        __builtin_prefetch(W2 + tid * 16, 0, 1);
        __builtin_prefetch(Wk + tid * 16, 0, 1);
    }

    // Phase 1: stage activation plane in LDS (padding already zero in Ain)
#if defined(USE_TDM)
    if (tid < 32) {
        // Tensor DMA descriptor (CDNA5 ISA ch8): contiguous 7168-elem x 1 row
        // 4-byte tile, global -> LDS, no multicast/iterate/pad.
        const unsigned long long ga =
            (unsigned long long)(uintptr_t)(const void*)(Ain + (size_t)n * PLANE);
        const unsigned int la = (unsigned int)(uintptr_t)(void*)Alds;
        u32x4 g0 = {
            1u,                                           // count=1, user D#
            la,                                           // lds_addr
            (unsigned int)ga,                             // global_addr[31:0]
            ((unsigned int)(ga >> 32) & 0x01FFFFFFu) |    // global_addr[56:32]
                (2u << 30)                                // type=2 ("image")
        };
        i32x8 g1 = {
            (int)(2u << 16),        // wg_mask=0, data_size=2 (4B)
            (int)(0x1C000000u),     // tensor_dim0[15:0]=7168 @ [63:48]
            (int)(0x00010000u),     // tensor_dim0[31:16]=0, tensor_dim1[15:0]=1
            (int)(0x1C000000u),     // tensor_dim1[31:16]=0, tile_dim0=7168
            (int)(0x00000001u),     // tile_dim1=1, tile_dim2=0
            (int)(0x00001C00u),     // tensor_dim0_stride[31:0]=7168
            (int)(0x1C000000u),     // stride0[47:32]=0, stride1[15:0]=7168
            0                       // stride1[47:16]=0
        };
        i32x4 z4 = {0, 0, 0, 0};
#if __clang_major__ >= 23
        i32x8 z8 = {0, 0, 0, 0, 0, 0, 0, 0};
        __builtin_amdgcn_tensor_load_to_lds(g0, g1, z4, z4, z8, 0);
#else
        __builtin_amdgcn_tensor_load_to_lds(g0, g1, z4, z4, 0);
#endif
        __builtin_amdgcn_s_wait_tensorcnt(0);
    }
#else
    for (int i = tid; i < PLANE; i += 256) Alds[i] = Ain[n * PLANE + i];
#endif
    if (tid < CH) { red1[tid] = 0.f; red2[tid] = 0.f; }
    __syncthreads();

    // Phase 2: 16x16 output tiles, K-loop of f32 WMMAs
    const int nOT = dout >> 4;      // 4 or 2 row-tiles
    const int T   = nOT * 7;        // x 7 column tiles (112/16)
    for (int t = wave; t < T; t += 8) {
        const int ot  = t / 7;
        const int lt  = t % 7;
        const int row = ot * 16 + lrow;        // A-frag: M = lane&15
        const int col = lt * 16 + lrow;        // B-frag: N = lane&15
        const int wb  = row * din + 2 * g;     // A-frag: K = j + 2g

        v8f acc1 = {0,0,0,0,0,0,0,0};
        v8f accV = {0,0,0,0,0,0,0,0};
        v8f accK = {0,0,0,0,0,0,0,0};
        for (int kb = 0; kb < din; kb += 4) {
            const int ko = kb + 2 * g;         // B-frag: K = j + 2g
            v2f b;  b.x  = Alds[ko * LP + col];
                    b.y  = Alds[(ko + 1) * LP + col];
            v2f a1; a1.x = W1[wb + kb]; a1.y = W1[wb + kb + 1];
            v2f a2; a2.x = W2[wb + kb]; a2.y = W2[wb + kb + 1];
            v2f ak; ak.x = Wk[wb + kb]; ak.y = Wk[wb + kb + 1];
            acc1 = wmma_f32(a1, b, acc1);
            accV = wmma_f32(a2, b, accV);
            accK = wmma_f32(ak, b, accK);
        }
        // C/D layout: vgpr r -> M = ot*16 + r + 8g, N = col
        const int m0 = ot * 16 + g * 8;
        const int cc = lt * 16 + lrow;
        #pragma unroll
        for (int r = 0; r < 8; ++r) {
            Slds[(m0 + r) * LP + cc] = acc1[r] + accV[r];
            // padding columns hold exact zeros -> contribute nothing
            float p1 = fabsf(accK[r]) * accV[r];
            float p2 = accK[r] * accK[r];
            p1 += __shfl_xor(p1, 1, 32);  p2 += __shfl_xor(p2, 1, 32);
            p1 += __shfl_xor(p1, 2, 32);  p2 += __shfl_xor(p2, 2, 32);
            p1 += __shfl_xor(p1, 4, 32);  p2 += __shfl_xor(p2, 4, 32);
            p1 += __shfl_xor(p1, 8, 32);  p2 += __shfl_xor(p2, 8, 32);
            if (lrow == 0) {
                atomicAdd(&red1[m0 + r], p1);
                atomicAdd(&red2[m0 + r], p2);
            }
        }
    }
    __syncthreads();

    // Phase 3: Y[o] = red1 / sqrt(red2)
    if (tid < CH) {
        float p2 = red2[tid];
        red1[tid] = (p2 > 0.f) ? red1[tid] * rsqrtf(p2) : 0.f;
    }
    __syncthreads();

    // Phase 4: epilogue + BN statistics. 4 threads per channel.
    const int c  = tid >> 2;
    const int l0 = tid & 3;
    const float Yc = red1[c];
    float s1 = 0.f, s2 = 0.f;
    float* outp = Aout + n * PLANE;
    for (int l = l0; l < LVALID; l += 4) {
        float v = 0.f;
        if (c < dout) {
            v = Slds[c * LP + l] + 0.1f * Yc;
            if (relu_bn) v = fmaxf(v, 0.f);
        }
        outp[c * LP + l] = v;
        s1 += v; s2 += v * v;
    }
    if (relu_bn) {
        s1 += __shfl_xor(s1, 1, 32);  s2 += __shfl_xor(s2, 1, 32);
        s1 += __shfl_xor(s1, 2, 32);  s2 += __shfl_xor(s2, 2, 32);
        if (l0 == 0 && c < dout) {
            atomicAdd(&gstats[c],      s1);
            atomicAdd(&gstats[CH + c], s2);
        }
    }
    // keep L-padding strip at exact zero for the next layer
    for (int i = tid; i < CH * (LP - LVALID); i += 256)
        outp[(i / (LP - LVALID)) * LP + LVALID + (i % (LP - LVALID))] = 0.f;
}

// ---------------------------------------------------------------------------
// BatchNorm apply (training-mode biased stats over N*L = 6400 per channel)
// ---------------------------------------------------------------------------
__global__ void bn_kernel(float* __restrict__ buf,
                          const float* __restrict__ gamma,
                          const float* __restrict__ beta,
                          const float* __restrict__ stats, int dout) {
    int idx = blockIdx.x * 256 + threadIdx.x;
    int total = N_EFF * dout * LVALID;
    if (idx >= total) return;
    int n   = idx / (dout * LVALID);
    int rem = idx % (dout * LVALID);
    int c   = rem / LVALID;
    int l   = rem % LVALID;
    const float inv_cnt = 1.0f / (N_EFF * LVALID);
    float mean  = stats[c] * inv_cnt;
    float var   = stats[CH + c] * inv_cnt - mean * mean;
    float scale = gamma[c] * rsqrtf(var + BN_EPS);
    float* p = buf + n * PLANE + c * LP + l;
    *p = (*p - mean) * scale + beta[c];
}

// ---------------------------------------------------------------------------
// Final per-(n,l,k) power normalization + layout to [B,nR,L,K,U,2]
// ---------------------------------------------------------------------------
__global__ void final_kernel(const float* __restrict__ buf,
                             float* __restrict__ out) {
    int idx = blockIdx.x * 256 + threadIdx.x;     // N * L * K
    if (idx >= N_EFF * LVALID * 8) return;
    int n = idx / (LVALID * 8);
    int l = (idx / 8) % LVALID;
    int k = idx % 8;
    const float* p = buf + n * PLANE;
    float fr0 = p[(2 * k + 0) * LP + l];
    float fr1 = p[(2 * k + 1) * LP + l];
    float fi0 = p[(16 + 2 * k + 0) * LP + l];
    float fi1 = p[(16 + 2 * k + 1) * LP + l];
    float inv = rsqrtf(fr0 * fr0 + fr1 * fr1 + fi0 * fi0 + fi1 * fi1);
    float* o = out + ((n * LVALID + l) * 8 + k) * 4;   // U*2 = 4
    o[0] = fr0 * inv;  o[1] = fi0 * inv;
    o[2] = fr1 * inv;  o[3] = fi1 * inv;
}

// ---------------------------------------------------------------------------
extern "C" void kernel_launch(void* const* d_in, const int* in_sizes, int n_in,
                              void* d_out, int out_size, void* d_ws, size_t ws_size,
                              hipStream_t stream) {
    (void)in_sizes; (void)n_in; (void)out_size; (void)ws_size;
    const float* re    = (const float*)d_in[0];
    const float* im    = (const float*)d_in[1];
    const float* P1_0  = (const float*)d_in[2];
    const float* P2_0  = (const float*)d_in[3];
    const float* k_0   = (const float*)d_in[5];   // q_* (d_in[4],[8],[12]) are dead
    const float* P1_1  = (const float*)d_in[6];
    const float* P2_1  = (const float*)d_in[7];
    const float* k_1   = (const float*)d_in[9];
    const float* P1_2  = (const float*)d_in[10];
    const float* P2_2  = (const float*)d_in[11];
    const float* k_2   = (const float*)d_in[13];
    const float* gam0  = (const float*)d_in[14];
    const float* bet0  = (const float*)d_in[15];
    const float* gam1  = (const float*)d_in[16];
    const float* bet1  = (const float*)d_in[17];

    float* ws    = (float*)d_ws;                  // needs ~3.7 MB
    float* bufA  = ws;
    float* bufB  = ws + (size_t)N_EFF * PLANE;
    float* stats = ws + (size_t)2 * N_EFF * PLANE;
    float* out   = (float*)d_out;

    const int packBlocks = (N_EFF * PLANE + 255) / 256;
    const int bnBlocks   = (N_EFF * CH * LVALID + 255) / 256;
    const int finBlocks  = (N_EFF * LVALID * 8 + 255) / 256;

    pack_kernel<<<packBlocks, 256, 0, stream>>>(re, im, bufA);

    zero_stats_kernel<<<1, 256, 0, stream>>>(stats);
    layer_kernel<<<N_EFF, 256, 0, stream>>>(bufA, bufB, P1_0, P2_0, k_0, 32, 64, stats, 1);
    bn_kernel<<<bnBlocks, 256, 0, stream>>>(bufB, gam0, bet0, stats, 64);

    zero_stats_kernel<<<1, 256, 0, stream>>>(stats);
    layer_kernel<<<N_EFF, 256, 0, stream>>>(bufB, bufA, P1_1, P2_1, k_1, 64, 64, stats, 1);
    bn_kernel<<<bnBlocks, 256, 0, stream>>>(bufA, gam1, bet1, stats, 64);

    layer_kernel<<<N_EFF, 256, 0, stream>>>(bufA, bufB, P1_2, P2_2, k_2, 64, 32, stats, 0);

    final_kernel<<<finBlocks, 256, 0, stream>>>(bufB, out);
}